// SAGEConv_34102040330288
// MI455X (gfx1250) — compile-verified
//
#include <hip/hip_runtime.h>

typedef __attribute__((ext_vector_type(2))) float v2f;
typedef __attribute__((ext_vector_type(8))) float v8f;

#define N_NODES 50000
#define N_EDGES 800000
#define D 128

// ---------------------------------------------------------------------------
// Kernel 1: zero the scratch accumulator [N_NODES x D] and degree [N_NODES]
// ---------------------------------------------------------------------------
__global__ void sage_zero_ws(float* __restrict__ acc, float* __restrict__ deg) {
    int tid    = blockIdx.x * blockDim.x + threadIdx.x;
    int stride = gridDim.x * blockDim.x;
    const int total = N_NODES * D;
    for (int j = tid; j < total; j += stride) acc[j] = 0.0f;
    for (int j = tid; j < N_NODES; j += stride) deg[j] = 0.0f;
}

// ---------------------------------------------------------------------------
// Kernel 2: edge scatter-add. One wave32 per edge; each lane handles 4 of the
// 128 feature channels with coalesced f32 atomics (accumulator is L2-resident:
// 25.6 MB << 192 MB L2).
// ---------------------------------------------------------------------------
__global__ void sage_scatter(const float* __restrict__ feat,
                             const int*   __restrict__ src,
                             const int*   __restrict__ dst,
                             float* __restrict__ acc,
                             float* __restrict__ deg) {
    int wave   = (blockIdx.x * blockDim.x + threadIdx.x) >> 5;
    int lane   = threadIdx.x & 31;
    int nwaves = (gridDim.x * blockDim.x) >> 5;

    for (int e = wave; e < N_EDGES; e += nwaves) {
        int s = src[e];             // wave-uniform -> scalarized by compiler
        int d = dst[e];
        const float* __restrict__ fr = feat + (long)s * D;
        float*       __restrict__ ar = acc  + (long)d * D;
#pragma unroll
        for (int c = 0; c < 4; ++c) {
            atomicAdd(&ar[lane + 32 * c], fr[lane + 32 * c]);
        }
        if (lane == 0) atomicAdd(&deg[d], 1.0f);
    }
}

// ---------------------------------------------------------------------------
// Kernel 3: fused WMMA GEMM:
//   out = feat @ W_self + (acc * 1/max(deg,1)) @ W_neigh + b_neigh
// One wave per 16x16 output tile; V_WMMA_F32_16X16X4_F32, K=128 in 32 steps.
// Grid sized so wave count == tile count exactly (EXEC all-ones for WMMA).
// ---------------------------------------------------------------------------
__global__ void __launch_bounds__(128)
sage_gemm(const float* __restrict__ feat,
          const float* __restrict__ acc,
          const float* __restrict__ deg,
          const float* __restrict__ Wself,
          const float* __restrict__ Wneigh,
          const float* __restrict__ bias,
          float* __restrict__ out) {
    const int gw = (blockIdx.x * blockDim.x + threadIdx.x) >> 5; // global wave id
    const int row_tile = gw >> 3;   // 3125 row tiles
    const int col_tile = gw & 7;    // 8 col tiles

    const int lane = threadIdx.x & 31;
    const int l16  = lane & 15;
    const int half = lane >> 4;
    const int kk   = half * 2;      // K sub-offset for this lane half

    const int row0 = row_tile * 16;
    const int col0 = col_tile * 16;
    const int arow = row0 + l16;    // A-matrix row owned by this lane
    const int bcol = col0 + l16;    // B-matrix col owned by this lane

    // fold mean normalization into the A operand of the second GEMM
    const float inv_deg = 1.0f / fmaxf(deg[arow], 1.0f);

    // C init = bias broadcast (N = lane&15 for every accumulator VGPR)
    const float bv = bias[bcol];
    v8f c;
#pragma unroll
    for (int i = 0; i < 8; ++i) c[i] = bv;

    const float* __restrict__ fA = feat + (long)arow * D;
    const float* __restrict__ nA = acc  + (long)arow * D;

    // GEMM 1: feat @ W_self
#pragma unroll 8
    for (int k = 0; k < D; k += 4) {
        v2f a = *(const v2f*)(fA + k + kk);          // A[l16][k+kk], [k+kk+1]
        v2f b;
        b.x = Wself[(k + kk) * D + bcol];            // B[k+kk][bcol]
        b.y = Wself[(k + kk + 1) * D + bcol];
        c = __builtin_amdgcn_wmma_f32_16x16x4_f32(
                false, a, false, b, (short)0, c, false, false);
    }

    // GEMM 2: (acc * inv_deg) @ W_neigh
#pragma unroll 8
    for (int k = 0; k < D; k += 4) {
        v2f a = *(const v2f*)(nA + k + kk);
        a.x *= inv_deg;
        a.y *= inv_deg;
        v2f b;
        b.x = Wneigh[(k + kk) * D + bcol];
        b.y = Wneigh[(k + kk + 1) * D + bcol];
        c = __builtin_amdgcn_wmma_f32_16x16x4_f32(
                false, a, false, b, (short)0, c, false, false);
    }

    // D store: VGPR i -> row i (lanes 0-15) / row 8+i (lanes 16-31), col = l16
#pragma unroll
    for (int i = 0; i < 8; ++i) {
        int m = half ? (8 + i) : i;
        out[(long)(row0 + m) * D + col0 + l16] = c[i];
    }
}

// ---------------------------------------------------------------------------
extern "C" void kernel_launch(void* const* d_in, const int* in_sizes, int n_in,
                              void* d_out, int out_size, void* d_ws, size_t ws_size,
                              hipStream_t stream) {
    const float* feat   = (const float*)d_in[0];
    const int*   src    = (const int*)  d_in[1];
    const int*   dst    = (const int*)  d_in[2];
    const float* Wself  = (const float*)d_in[3];
    const float* Wneigh = (const float*)d_in[4];
    const float* bias   = (const float*)d_in[5];
    float*       out    = (float*)d_out;

    float* acc = (float*)d_ws;                    // [N_NODES * D]
    float* deg = acc + (size_t)N_NODES * D;       // [N_NODES]

    sage_zero_ws<<<2048, 256, 0, stream>>>(acc, deg);
    sage_scatter<<<4096, 256, 0, stream>>>(feat, src, dst, acc, deg);

    // 3125 row tiles * 8 col tiles = 25000 waves; 4 waves/block -> 6250 blocks
    sage_gemm<<<6250, 128, 0, stream>>>(feat, acc, deg, Wself, Wneigh, bias, out);
}